// GAT_43593918054566
// MI455X (gfx1250) — compile-verified
//
#include <hip/hip_runtime.h>

#define F_IN 256
#define HID 128
#define NEG_SLOPE 0.2f
#define MBLK 32
#define LDSX (F_IN + 4) // padded LDS stride: 260 % 64 = 4 -> conflict-free

typedef float v2f __attribute__((ext_vector_type(2)));
typedef float v8f __attribute__((ext_vector_type(8)));

__device__ __forceinline__ float lrelu(float v) { return v > 0.f ? v : NEG_SLOPE * v; }
// order-preserving float<->int map so atomicMax(int) implements float max
__device__ __forceinline__ int enc_f(float f) {
    int b = __float_as_int(f);
    return b >= 0 ? b : (b ^ 0x7FFFFFFF);
}
__device__ __forceinline__ float dec_f(int b) {
    return __int_as_float(b >= 0 ? b : (b ^ 0x7FFFFFFF));
}
__device__ __forceinline__ void atomAddF(float* p, float v) {
#if defined(__HIP_DEVICE_COMPILE__)
    unsafeAtomicAdd(p, v); // guarantee global_atomic_add_f32 (no CAS loop)
#else
    atomicAdd(p, v);
#endif
}

// -------------------- K1: xw = x @ W via V_WMMA_F32_16X16X4_F32 --------------------
// 256 threads = 8 waves. Block tile: 32 rows x 128 cols.
// wave (mw in 0..1, nw in 0..3): rows [mw*16,+16), cols [nw*32,+32) = two 16x16 WMMA tiles.
__global__ __launch_bounds__(256) void gat_gemm(const float* __restrict__ x,
                                                const float* __restrict__ W,
                                                float* __restrict__ xw, int N) {
    __shared__ float xs[MBLK * LDSX];
    const int tid = threadIdx.x;
    const int rowBase = blockIdx.x * MBLK;
    const bool fullTile = (rowBase + MBLK) <= N;

    // cooperative load of 32x256 f32 tile (float4, coalesced), row-clamped for safety
    for (int i = tid; i < MBLK * (F_IN / 4); i += 256) {
        int r = i >> 6;      // 64 float4 per row
        int c4 = i & 63;
        int gr = rowBase + r;
        if (gr >= N) gr = N - 1;
        const float4 v = ((const float4*)(x + (size_t)gr * F_IN))[c4];
        float* dp = &xs[r * LDSX + c4 * 4];
        dp[0] = v.x; dp[1] = v.y; dp[2] = v.z; dp[3] = v.w;
    }
    __syncthreads();

    const int lane = tid & 31;
    const int wid  = tid >> 5;
    const int nw   = wid & 3;
    const int mw   = wid >> 2;
    const int col0 = nw * 32 + (lane & 15);
    const int koff = (lane >> 4) << 1;                 // lanes 16-31 hold K=+2,+3
    const float* xrow = &xs[(mw * 16 + (lane & 15)) * LDSX];

    v8f c0 = {}; v8f c1 = {};
    #pragma unroll 4
    for (int k0 = 0; k0 < F_IN; k0 += 4) {
        v2f a, b0, b1;
        a.x = xrow[k0 + koff];
        a.y = xrow[k0 + koff + 1];
        const float* wp = &W[(size_t)(k0 + koff) * HID + col0];
        b0.x = wp[0];    b0.y = wp[HID];
        b1.x = wp[16];   b1.y = wp[HID + 16];
        c0 = __builtin_amdgcn_wmma_f32_16x16x4_f32(false, a, false, b0, (short)0, c0, false, false);
        c1 = __builtin_amdgcn_wmma_f32_16x16x4_f32(false, a, false, b1, (short)0, c1, false, false);
    }

    // C/D layout: VGPR r -> M=r (lanes 0-15), M=r+8 (lanes 16-31); N = lane&15
    const int mBase = rowBase + mw * 16 + ((lane >> 4) << 3);
    float* op = &xw[(size_t)mBase * HID + col0];
    if (fullTile) {
        // one branch for the whole tile; 16 unguarded stores
        #pragma unroll
        for (int r = 0; r < 8; ++r) {
            op[(size_t)r * HID]      = c0[r];
            op[(size_t)r * HID + 16] = c1[r];
        }
    } else {
        #pragma unroll
        for (int r = 0; r < 8; ++r) {
            if (mBase + r < N) {
                op[(size_t)r * HID]      = c0[r];
                op[(size_t)r * HID + 16] = c1[r];
            }
        }
    }
}

// -------------------- K2: per-node a_s, a_d; init emax with self-loop score ---------
// one wave per node; one float4 per lane covers the 128-float row exactly (b128 load)
__global__ __launch_bounds__(256) void gat_node_att(const float* __restrict__ xw,
                                                    const float* __restrict__ att_src,
                                                    const float* __restrict__ att_dst,
                                                    float* __restrict__ a_s,
                                                    float* __restrict__ a_d,
                                                    int* __restrict__ emax_enc, int N) {
    const int node = (blockIdx.x * 256 + threadIdx.x) >> 5;
    const int lane = threadIdx.x & 31;
    if (node >= N) return;
    const float4 v  = ((const float4*)(xw + (size_t)node * HID))[lane];
    const float4 as = ((const float4*)att_src)[lane];
    const float4 ad = ((const float4*)att_dst)[lane];
    float s = v.x * as.x + v.y * as.y + v.z * as.z + v.w * as.w;
    float d = v.x * ad.x + v.y * ad.y + v.z * ad.z + v.w * ad.w;
    #pragma unroll
    for (int off = 16; off > 0; off >>= 1) {
        s += __shfl_xor(s, off, 32);
        d += __shfl_xor(d, off, 32);
    }
    if (lane == 0) {
        a_s[node] = s;
        a_d[node] = d;
        emax_enc[node] = enc_f(lrelu(s + d)); // self-loop seeds the segment max
    }
}

// -------------------- K3: segment max over edges (atomic max on int-encoded f32) ----
__global__ void gat_edge_max(const int* __restrict__ ei, const float* __restrict__ a_s,
                             const float* __restrict__ a_d, int* __restrict__ emax_enc, int E) {
    const int e = blockIdx.x * blockDim.x + threadIdx.x;
    if (e >= E) return;
    const int s = ei[e], d = ei[E + e];
    atomicMax(&emax_enc[d], enc_f(lrelu(a_s[s] + a_d[d])));
}

// -------------------- K4: denom = self-loop exp term (plain store, pre-atomics) -----
__global__ void gat_node_denom(const float* __restrict__ a_s, const float* __restrict__ a_d,
                               const int* __restrict__ emax_enc, float* __restrict__ denom, int N) {
    const int i = blockIdx.x * blockDim.x + threadIdx.x;
    if (i >= N) return;
    denom[i] = __expf(lrelu(a_s[i] + a_d[i]) - dec_f(emax_enc[i]));
}

// -------------------- K5: denom += exp(e - emax) over edges -------------------------
__global__ void gat_edge_denom(const int* __restrict__ ei, const float* __restrict__ a_s,
                               const float* __restrict__ a_d, const int* __restrict__ emax_enc,
                               float* __restrict__ denom, int E) {
    const int e = blockIdx.x * blockDim.x + threadIdx.x;
    if (e >= E) return;
    const int s = ei[e], d = ei[E + e];
    atomAddF(&denom[d], __expf(lrelu(a_s[s] + a_d[d]) - dec_f(emax_enc[d])));
}

// -------------------- K6: out = coef_self * xw (initializes d_out), float4 ----------
__global__ void gat_self_out(const float* __restrict__ xw, const float* __restrict__ a_s,
                             const float* __restrict__ a_d, const int* __restrict__ emax_enc,
                             const float* __restrict__ denom, float* __restrict__ out, int total4) {
    const int idx4 = blockIdx.x * blockDim.x + threadIdx.x;
    if (idx4 >= total4) return;
    const int i = idx4 >> 5; // 32 float4 per node row
    const float coef = __expf(lrelu(a_s[i] + a_d[i]) - dec_f(emax_enc[i])) / denom[i];
    float4 v = ((const float4*)xw)[idx4];
    v.x *= coef; v.y *= coef; v.z *= coef; v.w *= coef;
    ((float4*)out)[idx4] = v;
}

// -------------------- K7: one wave per edge: out[dst] += coef * xw[src] -------------
// one b128 gather per lane, 4 f32 hardware atomics to consecutive addresses
__global__ __launch_bounds__(256) void gat_edge_agg(const int* __restrict__ ei,
                                                    const float* __restrict__ xw,
                                                    const float* __restrict__ a_s,
                                                    const float* __restrict__ a_d,
                                                    const int* __restrict__ emax_enc,
                                                    const float* __restrict__ denom,
                                                    float* __restrict__ out, int E) {
    const int e = (blockIdx.x * 256 + threadIdx.x) >> 5;
    const int lane = threadIdx.x & 31;
    if (e >= E) return;
    const int s = ei[e], d = ei[E + e];
    const float coef = __expf(lrelu(a_s[s] + a_d[d]) - dec_f(emax_enc[d])) / denom[d];
    const float4 v = ((const float4*)(xw + (size_t)s * HID))[lane];
    float* dstp = out + (size_t)d * HID + lane * 4;
    atomAddF(dstp + 0, coef * v.x);
    atomAddF(dstp + 1, coef * v.y);
    atomAddF(dstp + 2, coef * v.z);
    atomAddF(dstp + 3, coef * v.w);
}

// -------------------- K8: out = relu(out + bias), float4 ----------------------------
__global__ void gat_finalize(float* __restrict__ out, const float* __restrict__ bias, int total4) {
    const int idx4 = blockIdx.x * blockDim.x + threadIdx.x;
    if (idx4 >= total4) return;
    const float4 b = ((const float4*)bias)[idx4 & 31];
    float4 v = ((float4*)out)[idx4];
    v.x += b.x; v.y += b.y; v.z += b.z; v.w += b.w;
    v.x = v.x > 0.f ? v.x : 0.f;
    v.y = v.y > 0.f ? v.y : 0.f;
    v.z = v.z > 0.f ? v.z : 0.f;
    v.w = v.w > 0.f ? v.w : 0.f;
    ((float4*)out)[idx4] = v;
}

extern "C" void kernel_launch(void* const* d_in, const int* in_sizes, int n_in,
                              void* d_out, int out_size, void* d_ws, size_t ws_size,
                              hipStream_t stream) {
    const float* x       = (const float*)d_in[0];
    const int*   ei      = (const int*)d_in[1];
    const float* W       = (const float*)d_in[2];
    const float* att_src = (const float*)d_in[3];
    const float* att_dst = (const float*)d_in[4];
    const float* bias    = (const float*)d_in[5];
    float* out = (float*)d_out;

    const int N = in_sizes[0] / F_IN;
    const int E = in_sizes[1] / 2;
    const int total4 = (N * HID) / 4;

    // workspace carve-up (fully rewritten every call before any read)
    float* xw      = (float*)d_ws;
    float* a_s     = xw + (size_t)N * HID;
    float* a_d     = a_s + N;
    int*   emax    = (int*)(a_d + N);
    float* denom   = (float*)(emax + N);

    gat_gemm      <<<(N + MBLK - 1) / MBLK, 256, 0, stream>>>(x, W, xw, N);
    gat_node_att  <<<(N + 7) / 8,           256, 0, stream>>>(xw, att_src, att_dst, a_s, a_d, emax, N);
    gat_edge_max  <<<(E + 255) / 256,       256, 0, stream>>>(ei, a_s, a_d, emax, E);
    gat_node_denom<<<(N + 255) / 256,       256, 0, stream>>>(a_s, a_d, emax, denom, N);
    gat_edge_denom<<<(E + 255) / 256,       256, 0, stream>>>(ei, a_s, a_d, emax, denom, E);
    gat_self_out  <<<(total4 + 255) / 256,  256, 0, stream>>>(xw, a_s, a_d, emax, denom, out, total4);
    gat_edge_agg  <<<(E + 7) / 8,           256, 0, stream>>>(ei, xw, a_s, a_d, emax, denom, out, E);
    gat_finalize  <<<(total4 + 255) / 256,  256, 0, stream>>>(out, bias, total4);
}